// FrequencyAttention_60773787238682
// MI455X (gfx1250) — compile-verified
//
#include <hip/hip_runtime.h>
#include <math.h>

// ---------------------------------------------------------------------------
// CDNA5 (gfx1250) wave32 WMMA fp32 pipeline for FrequencyAttention.
// - All GEMM-shaped work on V_WMMA_F32_16X16X4_F32
// - conv im2col staging: global_load_async_to_lds_b128 + s_wait_asynccnt
// - DFT matrix staging: TDM tensor_load_to_lds with HW padding + s_wait_tensorcnt
// Fragment layouts (per CDNA5 ISA):
//   A (16x4,  MxK): vgpr r, lane L: M = L%16, K = 2*(L/16)+r
//   B (4x16,  KxN): vgpr r, lane L: N = L%16, K = 2*(L/16)+r
//   C/D (16x16):    vgpr r, lane L: M = r+8*(L/16), N = L%16
// ---------------------------------------------------------------------------

typedef float v2f __attribute__((ext_vector_type(2)));
typedef float v4f __attribute__((ext_vector_type(4)));
typedef float v8f __attribute__((ext_vector_type(8)));
typedef int v4i __attribute__((vector_size(16)));
typedef unsigned int v4u __attribute__((vector_size(16)));
typedef int v8i __attribute__((vector_size(32)));

#define DEV static __device__ __forceinline__
#define PI_F 3.14159265358979323846f
#define BN_EPS 1e-5f

// ---- gfx1250 DMA paths (guarded; fall back to plain ds stores) ----
#define TRY_ASYNC 1
#define TRY_TDM 1
#if defined(__has_builtin)
#if TRY_ASYNC && __has_builtin(__builtin_amdgcn_global_load_async_to_lds_b128)
#define HAVE_ASYNC_LDS 1
#endif
#if __has_builtin(__builtin_amdgcn_s_wait_asynccnt)
#define HAVE_WAIT_ASYNC 1
#endif
#if TRY_TDM && __has_builtin(__builtin_amdgcn_tensor_load_to_lds) && \
    __has_builtin(__builtin_amdgcn_s_wait_tensorcnt)
#define HAVE_TDM 1
#endif
#endif

DEV void lds_copy16(const float* __restrict__ g, float* __restrict__ l) {
#if defined(HAVE_ASYNC_LDS)
  // proto: (v4i addrspace(1)*, v4i addrspace(3)*, imm off, imm cpol).
  // flat->AS1 is identity; flat->AS3 truncation gives the LDS offset
  // (ISA 10.2: the LDS address is the low 32 bits).
  typedef __attribute__((address_space(1))) v4i* gp_t;
  typedef __attribute__((address_space(3))) v4i* lp_t;
#pragma unroll
  for (int i = 0; i < 16; i += 4)
    __builtin_amdgcn_global_load_async_to_lds_b128(
        (gp_t)(unsigned long long)(g + i),
        (lp_t)(unsigned int)(unsigned long long)(l + i), 0, 0);
#else
#pragma unroll
  for (int i = 0; i < 16; i += 4) *(v4f*)(l + i) = *(const v4f*)(g + i);
#endif
}

DEV void lds_copy_fence() {
#if defined(HAVE_ASYNC_LDS) && defined(HAVE_WAIT_ASYNC)
  __builtin_amdgcn_s_wait_asynccnt(0);
#endif
  __syncthreads();
}

#if defined(HAVE_TDM)
// TDM load of a contiguous 64x64 f32 tile into LDS with hardware padding:
// pad_interval=5 (every 64 DWORDs) + pad_amount=3 (4 DWORDs) reproduces the
// 68-float padded LDS row stride.  D# per ISA 8.3/8.4.
DEV void tdm_load_tile64(const float* gsrc, float* ldst) {
  const unsigned long long ga = (unsigned long long)gsrc;
  const unsigned int lofs = (unsigned int)(unsigned long long)ldst;
  v4u g0 = {1u,                                    // count=1, user desc
            lofs,                                  // lds_addr [63:32]
            (unsigned int)(ga & 0xFFFFFFFFu),      // global_addr [95:64]
            (unsigned int)(((ga >> 32) & 0x01FFFFFFu) | (2u << 30))}; // type=2
  v8i g1 = {(int)((2u << 16) | (1u << 20) | (5u << 22) | (3u << 25)),
            // ^ data_size=4B, pad_enable, pad_interval=64dw, pad_amount=4dw
            (int)(64u << 16),                      // tensor_dim0 = 64
            (int)(64u << 16),                      // tensor_dim1 = 64
            (int)(64u << 16),                      // tile_dim0   = 64
            64,                                    // tile_dim1   = 64
            64,                                    // tensor_dim0_stride = 64
            0, 0};
  v4i g2 = {0, 0, 0, 0};
  v4i g3 = {0, 0, 0, 0};
#if __clang_major__ >= 23
  v8i g4 = {0, 0, 0, 0, 0, 0, 0, 0};
  __builtin_amdgcn_tensor_load_to_lds(g0, g1, g2, g3, g4, 0);
#else
  __builtin_amdgcn_tensor_load_to_lds(g0, g1, g2, g3, 0);
#endif
}
#endif

DEV v8f wmma4(v2f a, v2f b, v8f c) {
  return __builtin_amdgcn_wmma_f32_16x16x4_f32(false, a, false, b, (short)0, c,
                                               false, false);
}

// complex C += A*B (f32 WMMA has no A/B neg bits -> negate imag frag in VALU)
DEV void cmma(v2f ar, v2f ai, v2f br, v2f bi, v8f& cre, v8f& cim) {
  cre = wmma4(ar, br, cre);
  v2f nai; nai[0] = -ai[0]; nai[1] = -ai[1];
  cre = wmma4(nai, bi, cre);
  cim = wmma4(ar, bi, cim);
  cim = wmma4(ai, br, cim);
}

// ===========================================================================
// 0) one-shot weight repack: w[(o*512+k)*9+tap] -> wr[((tap*256+o)*512)+k]
// ===========================================================================
__global__ void k_repack_w(const float* __restrict__ w, float* __restrict__ wr) {
  const size_t i = (size_t)blockIdx.x * 256 + threadIdx.x;  // 9*256*512
  const int tap = (int)(i >> 17);
  const int rem = (int)(i & 131071);
  const int o = rem >> 9, k = rem & 511;
  wr[i] = w[((size_t)o * 512 + k) * 9 + tap];
}

// ===========================================================================
// 1) dilated 3x3 conv (512->256, dil=3, pad=3) + bias + BN + ReLU.
// ===========================================================================
__global__ void __launch_bounds__(512)
k_conv_bn_relu(const float* __restrict__ x,
               const float* __restrict__ wr,
               const float* __restrict__ cb,
               const float* __restrict__ bg, const float* __restrict__ bb,
               const float* __restrict__ bm, const float* __restrict__ bv,
               float* __restrict__ out) {
  __shared__ float lx[16 * 516];             // padded row stride: 4-bank skew
  const int lane = threadIdx.x & 31, wid = threadIdx.x >> 5;   // 16 waves
  const int half = lane >> 4, l16 = lane & 15;
  const int b  = blockIdx.x >> 8;
  const int Nb = (blockIdx.x & 255) << 4;
  const int Mb = wid << 4;
  const int p  = Nb + l16;

  const int spix = threadIdx.x >> 5;         // staging: pixel per wave-group
  const int sseg = (threadIdx.x & 31) << 4;  // 16 channels per thread
  const int spp  = Nb + spix;
  const int spy  = spp >> 6, spx = spp & 63;

  v8f acc = {};
  for (int tap = 0; tap < 9; ++tap) {
    const int yy = spy + (tap / 3) * 3 - 3;
    const int xx = spx + (tap % 3) * 3 - 3;
    const bool valid = ((unsigned)yy < 64u) && ((unsigned)xx < 64u);
    float* dst = &lx[spix * 516 + sseg];
    if (valid) {
      lds_copy16(x + (((size_t)b * 4096 + yy * 64 + xx) << 9) + sseg, dst);
    } else {
      v4f z = {};
#pragma unroll
      for (int i = 0; i < 16; i += 4) *(v4f*)(dst + i) = z;
    }
    lds_copy_fence();

    const float* wa = wr + ((size_t)(tap * 256 + Mb + l16) << 9);
#pragma unroll 4
    for (int kc = 0; kc < 512; kc += 4) {
      const int k = kc + 2 * half;
      const v2f a  = *(const v2f*)(wa + k);
      const v2f bf = *(const v2f*)&lx[l16 * 516 + k];
      acc = wmma4(a, bf, acc);
    }
    __syncthreads();                         // before next tap overwrites lx
  }
#pragma unroll
  for (int r = 0; r < 8; ++r) {
    const int oc = Mb + r + 8 * half;
    const float inv = bg[oc] * rsqrtf(bv[oc] + BN_EPS);
    float v = acc[r] + cb[oc];
    v = v * inv + (bb[oc] - bm[oc] * inv);
    out[(((size_t)b * 256 + oc) << 12) + p] = fmaxf(v, 0.f);
  }
}

// ===========================================================================
// 2) batched 64-point DFT as WMMA GEMM. One block = one 64x64 matrix staged
//    in LDS via TDM (stride-68 padding done by the DMA engine); 8 waves x 2
//    tiles.  left==1: out = W @ X; left==0: out = X @ W.
// ===========================================================================
__global__ void __launch_bounds__(256)
k_dft64(const float* __restrict__ in_re, const float* __restrict__ in_im,
        float* __restrict__ out_re, float* __restrict__ out_im,
        int left, float sign, float scale) {
  __shared__ float lre[64 * 68];
  __shared__ float lim[64 * 68];
  const size_t base = (size_t)blockIdx.x << 12;

#if defined(HAVE_TDM)
  if (threadIdx.x < 32) {                    // wave 0 drives the TDM
    tdm_load_tile64(in_re + base, lre);
    if (in_im) tdm_load_tile64(in_im + base, lim);
    __builtin_amdgcn_s_wait_tensorcnt(0);
  }
  __syncthreads();
#else
  {
    const int row = threadIdx.x >> 2;
    const int col = (threadIdx.x & 3) << 4;
    lds_copy16(in_re + base + row * 64 + col, &lre[row * 68 + col]);
    if (in_im) lds_copy16(in_im + base + row * 64 + col, &lim[row * 68 + col]);
  }
  lds_copy_fence();
#endif

  const int lane = threadIdx.x & 31, wid = threadIdx.x >> 5;
  const int half = lane >> 4, l16 = lane & 15;
  const float w0 = sign * (PI_F / 32.0f);

  for (int pass = 0; pass < 2; ++pass) {
    const int t  = wid + (pass << 3);
    const int Mb = (t >> 2) << 4, Nb = (t & 3) << 4;
    const int m = Mb + l16, n = Nb + l16;
    v8f cre = {}, cim = {};
    for (int kb = 0; kb < 64; kb += 4) {
      const int k = kb + 2 * half;
      v2f wrf, wif;
      float s0, c0, s1, c1;
      if (left) {
        __sincosf(w0 * (float)(m * k), &s0, &c0);
        __sincosf(w0 * (float)(m * (k + 1)), &s1, &c1);
        wrf[0] = c0; wrf[1] = c1; wif[0] = s0; wif[1] = s1;
        v2f dr, di;
        dr[0] = lre[k * 68 + n]; dr[1] = lre[(k + 1) * 68 + n];
        if (in_im) {
          di[0] = lim[k * 68 + n]; di[1] = lim[(k + 1) * 68 + n];
          cmma(wrf, wif, dr, di, cre, cim);
        } else {
          cre = wmma4(wrf, dr, cre);
          cim = wmma4(wif, dr, cim);
        }
      } else {
        __sincosf(w0 * (float)(k * n), &s0, &c0);
        __sincosf(w0 * (float)((k + 1) * n), &s1, &c1);
        wrf[0] = c0; wrf[1] = c1; wif[0] = s0; wif[1] = s1;
        const v2f dr = *(const v2f*)&lre[m * 68 + k];
        if (in_im) {
          const v2f di = *(const v2f*)&lim[m * 68 + k];
          cmma(dr, di, wrf, wif, cre, cim);
        } else {
          cre = wmma4(dr, wrf, cre);
          cim = wmma4(dr, wif, cim);
        }
      }
    }
#pragma unroll
    for (int r = 0; r < 8; ++r) {
      const size_t idx = base + (size_t)(Mb + r + 8 * half) * 64 + (Nb + l16);
      out_re[idx] = cre[r] * scale;
      out_im[idx] = cim[r] * scale;
    }
  }
}

// ===========================================================================
// 3) 32-point inverse DFT over per-head channel axis (left, K=32)
// ===========================================================================
__global__ void k_idft32_left(const float* __restrict__ in_re,
                              const float* __restrict__ in_im,
                              float* __restrict__ out_re,
                              float* __restrict__ out_im) {
  const int lane = threadIdx.x & 31, wid = threadIdx.x >> 5;
  const int half = lane >> 4, l16 = lane & 15;
  const int wt = blockIdx.x * 8 + wid;       // 32768
  const int g  = wt >> 9;
  const int t  = wt & 511;
  const int Mb = (t >> 8) << 4, Nb = (t & 255) << 4;
  const size_t base = (size_t)g * 32 * 4096;
  const float w0 = PI_F / 16.0f;             // +2*pi/32

  v8f cre = {}, cim = {};
  for (int kb = 0; kb < 32; kb += 4) {
    const int k = kb + 2 * half;
    const int m = Mb + l16, n = Nb + l16;
    v2f wrf, wif, dr, di;
    float s0, c0, s1, c1;
    __sincosf(w0 * (float)(m * k), &s0, &c0);
    __sincosf(w0 * (float)(m * (k + 1)), &s1, &c1);
    wrf[0] = c0; wrf[1] = c1; wif[0] = s0; wif[1] = s1;
    dr[0] = in_re[base + (size_t)k * 4096 + n];
    dr[1] = in_re[base + (size_t)(k + 1) * 4096 + n];
    di[0] = in_im[base + (size_t)k * 4096 + n];
    di[1] = in_im[base + (size_t)(k + 1) * 4096 + n];
    cmma(wrf, wif, dr, di, cre, cim);
  }
#pragma unroll
  for (int r = 0; r < 8; ++r) {
    const size_t idx = base + (size_t)(Mb + r + 8 * half) * 4096 + (Nb + l16);
    out_re[idx] = cre[r] * (1.0f / 32.0f);
    out_im[idx] = cim[r] * (1.0f / 32.0f);
  }
}

// four-step twiddle: element (row k1, col n2) *= exp(+2*pi*i*k1*n2/4096)
__global__ void k_twiddle4096(float* __restrict__ re, float* __restrict__ im) {
  const size_t idx = (size_t)blockIdx.x * 256 + threadIdx.x;
  const int ij = (int)(idx & 4095);
  const int i = ij >> 6, j = ij & 63;
  float s, c;
  __sincosf((2.0f * PI_F / 4096.0f) * (float)(i * j), &s, &c);
  const float r = re[idx], m = im[idx];
  re[idx] = r * c - m * s;
  im[idx] = r * s + m * c;
}

// ===========================================================================
// 4) attention
// ===========================================================================
__global__ void k_rownorm(const float* __restrict__ fr,
                          const float* __restrict__ fi,
                          float* __restrict__ norms) {
  const int row = blockIdx.x;
  const size_t base = (size_t)row << 12;
  float s = 0.f;
  for (int i = threadIdx.x; i < 4096; i += 256) {
    const float a = fr[base + i], b = fi[base + i];
    s += a * a + b * b;
  }
  __shared__ float red[256];
  red[threadIdx.x] = s;
  __syncthreads();
  for (int off = 128; off; off >>= 1) {
    if ((int)threadIdx.x < off) red[threadIdx.x] += red[threadIdx.x + off];
    __syncthreads();
  }
  if (threadIdx.x == 0) norms[row] = fmaxf(sqrtf(red[0]), 1e-12f);
}

// G[c,d] = sum_n f[c,n]*f[d,n] (complex, no conjugate)
__global__ void k_gram(const float* __restrict__ fr,
                       const float* __restrict__ fi,
                       float* __restrict__ gr, float* __restrict__ gi) {
  const int g = blockIdx.x;                  // 64 (b,head)
  const int lane = threadIdx.x & 31, wid = threadIdx.x >> 5;  // 4 waves
  const int half = lane >> 4, l16 = lane & 15;
  const int Mb = (wid >> 1) << 4, Nb = (wid & 1) << 4;
  const size_t base = (size_t)g * 32 * 4096;
  const size_t rm = base + (size_t)(Mb + l16) * 4096;
  const size_t rn = base + (size_t)(Nb + l16) * 4096;

  v8f cre = {}, cim = {};
  for (int kb = 0; kb < 4096; kb += 4) {
    const int k = kb + 2 * half;
    const v2f ar = *(const v2f*)&fr[rm + k];
    const v2f ai = *(const v2f*)&fi[rm + k];
    const v2f br = *(const v2f*)&fr[rn + k];
    const v2f bi = *(const v2f*)&fi[rn + k];
    cmma(ar, ai, br, bi, cre, cim);
  }
#pragma unroll
  for (int r = 0; r < 8; ++r) {
    const int idx = g * 1024 + (Mb + r + 8 * half) * 32 + (Nb + l16);
    gr[idx] = cre[r];
    gi[idx] = cim[r];
  }
}

__global__ void k_softmax(const float* __restrict__ gr,
                          const float* __restrict__ gi,
                          const float* __restrict__ norms,
                          const float* __restrict__ temp,
                          float* __restrict__ ar_, float* __restrict__ ai_) {
  const int g = blockIdx.x, head = g & 7;
  const int d = threadIdx.x, c = threadIdx.y;  // one wave == one row
  const float scale = temp[head] / (norms[g * 32 + c] * norms[g * 32 + d]);
  const int idx = g * 1024 + c * 32 + d;
  const float vr = gr[idx] * scale, vi = gi[idx] * scale;
  float mr = vr, mi = vi;
  for (int off = 16; off; off >>= 1) {
    mr = fmaxf(mr, __shfl_xor(mr, off, 32));
    mi = fmaxf(mi, __shfl_xor(mi, off, 32));
  }
  const float er = __expf(vr - mr), ei = __expf(vi - mi);
  float sr = er, si = ei;
  for (int off = 16; off; off >>= 1) {
    sr += __shfl_xor(sr, off, 32);
    si += __shfl_xor(si, off, 32);
  }
  ar_[idx] = er / sr;
  ai_[idx] = ei / si;
}

// out[c,n] = sum_d attn[c,d]*f[d,n] (complex, K=32)
__global__ void k_attn_apply(const float* __restrict__ ar_,
                             const float* __restrict__ ai_,
                             const float* __restrict__ fr,
                             const float* __restrict__ fi,
                             float* __restrict__ or_, float* __restrict__ oi_) {
  const int lane = threadIdx.x & 31, wid = threadIdx.x >> 5;
  const int half = lane >> 4, l16 = lane & 15;
  const int wt = blockIdx.x * 8 + wid;       // 32768
  const int g  = wt >> 9;
  const int t  = wt & 511;
  const int Mb = (t >> 8) << 4, Nb = (t & 255) << 4;
  const size_t base = (size_t)g * 32 * 4096;
  const int abase = g * 1024;

  v8f cre = {}, cim = {};
#pragma unroll
  for (int kb = 0; kb < 32; kb += 4) {
    const int k = kb + 2 * half;
    const int m = Mb + l16, n = Nb + l16;
    const v2f ar = *(const v2f*)&ar_[abase + m * 32 + k];
    const v2f ai = *(const v2f*)&ai_[abase + m * 32 + k];
    v2f br, bi;
    br[0] = fr[base + (size_t)k * 4096 + n];
    br[1] = fr[base + (size_t)(k + 1) * 4096 + n];
    bi[0] = fi[base + (size_t)k * 4096 + n];
    bi[1] = fi[base + (size_t)(k + 1) * 4096 + n];
    cmma(ar, ai, br, bi, cre, cim);
  }
#pragma unroll
  for (int r = 0; r < 8; ++r) {
    const size_t idx = base + (size_t)(Mb + r + 8 * half) * 4096 + (Nb + l16);
    or_[idx] = cre[r];
    oi_[idx] = cim[r];
  }
}

// ===========================================================================
// 5) gating branch
// ===========================================================================
__global__ void k_g1(const float* __restrict__ fr,
                     const float* __restrict__ w1,
                     const float* __restrict__ w1b,
                     const float* __restrict__ bg, const float* __restrict__ bb,
                     const float* __restrict__ bm, const float* __restrict__ bv,
                     float* __restrict__ g1) {
  const int lane = threadIdx.x & 31, wid = threadIdx.x >> 5;
  const int half = lane >> 4, l16 = lane & 15;
  const int wt = blockIdx.x * 8 + wid;       // 2048
  const int b  = wt >> 8;
  const int Nb = (wt & 255) << 4;
  const int p  = Nb + l16;

  v8f acc = {};
#pragma unroll 4
  for (int kb = 0; kb < 256; kb += 4) {
    const int k = kb + 2 * half;
    const v2f a = *(const v2f*)&w1[l16 * 256 + k];
    v2f bf;
    bf[0] = fr[(((size_t)b * 256 + k) << 12) + p];
    bf[1] = fr[(((size_t)b * 256 + k + 1) << 12) + p];
    acc = wmma4(a, bf, acc);
  }
#pragma unroll
  for (int r = 0; r < 8; ++r) {
    const int oc = r + 8 * half;
    const float inv = bg[oc] * rsqrtf(bv[oc] + BN_EPS);
    float v = acc[r] + w1b[oc];
    v = v * inv + (bb[oc] - bm[oc] * inv);
    g1[(((size_t)b * 16 + oc) << 12) + p] = fmaxf(v, 0.f);
  }
}

__global__ void k_gate(const float* __restrict__ g1,
                       const float* __restrict__ w2,
                       const float* __restrict__ w2b,
                       const float* __restrict__ fr,
                       const float* __restrict__ fi,
                       float* __restrict__ outr, float* __restrict__ outi) {
  const int lane = threadIdx.x & 31, wid = threadIdx.x >> 5;
  const int half = lane >> 4, l16 = lane & 15;
  const int wt = blockIdx.x * 8 + wid;       // 32768
  const int b  = wt >> 12;
  const int t  = wt & 4095;
  const int Mb = (t >> 8) << 4, Nb = (t & 255) << 4;
  const int p  = Nb + l16;

  v8f acc = {};
#pragma unroll
  for (int kb = 0; kb < 16; kb += 4) {
    const int k = kb + 2 * half;
    const v2f a = *(const v2f*)&w2[(Mb + l16) * 16 + k];
    v2f bf;
    bf[0] = g1[(((size_t)b * 16 + k) << 12) + p];
    bf[1] = g1[(((size_t)b * 16 + k + 1) << 12) + p];
    acc = wmma4(a, bf, acc);
  }
#pragma unroll
  for (int r = 0; r < 8; ++r) {
    const int oc = Mb + r + 8 * half;
    const size_t idx = (((size_t)b * 256 + oc) << 12) + p;
    const float gate = 1.0f / (1.0f + __expf(-(acc[r] + w2b[oc])));
    outr[idx] = gate * fr[idx];
    outi[idx] = gate * fi[idx];
  }
}

// ===========================================================================
// 6) magnitudes
// ===========================================================================
__global__ void k_abs(const float* __restrict__ re, const float* __restrict__ im,
                      float* __restrict__ out) {
  const size_t i = (size_t)blockIdx.x * 256 + threadIdx.x;
  out[i] = sqrtf(re[i] * re[i] + im[i] * im[i]);
}

__global__ void k_abs_remap(const float* __restrict__ re,
                            const float* __restrict__ im,
                            float* __restrict__ out) {
  const size_t i = (size_t)blockIdx.x * 256 + threadIdx.x;
  const size_t row = i >> 12;
  const int q = (int)(i & 4095);
  const int k1 = q >> 6, k2 = q & 63;
  out[(row << 12) + (size_t)(k1 + (k2 << 6))] =
      sqrtf(re[i] * re[i] + im[i] * im[i]);
}

// ===========================================================================
// 7) projection: out2 = concat(out_f2, out_l2) + x;  F2 = P @ out2 + pb
// ===========================================================================
__global__ void k_proj(const float* __restrict__ of2,
                       const float* __restrict__ ol2,
                       const float* __restrict__ x,
                       const float* __restrict__ pw,
                       const float* __restrict__ pb,
                       float* __restrict__ out) {
  const int lane = threadIdx.x & 31, wid = threadIdx.x >> 5;
  const int half = lane >> 4, l16 = lane & 15;
  const int wt = blockIdx.x * 8 + wid;       // 65536
  const int b  = wt >> 13;
  const int t  = wt & 8191;
  const int Mb = (t >> 8) << 4, Nb = (t & 255) << 4;
  const int p  = Nb + l16;
  const float* xrow = x + (((size_t)b * 4096 + p) << 9);

  v8f acc = {};
#pragma unroll 4
  for (int kb = 0; kb < 512; kb += 4) {
    const int k = kb + 2 * half;
    const v2f a  = *(const v2f*)&pw[(size_t)(Mb + l16) * 512 + k];
    const v2f xr = *(const v2f*)&xrow[k];
    const float* plane = (k < 256)
        ? &of2[(((size_t)b * 256 + k) << 12) + p]
        : &ol2[(((size_t)b * 256 + (k - 256)) << 12) + p];
    v2f bf;
    bf[0] = plane[0] + xr[0];
    bf[1] = plane[4096] + xr[1];
    acc = wmma4(a, bf, acc);
  }
#pragma unroll
  for (int r = 0; r < 8; ++r) {
    const int o = Mb + r + 8 * half;
    out[(((size_t)b * 4096 + p) << 9) + o] = acc[r] + pb[o];
  }
}

// ===========================================================================
// host launcher
// ===========================================================================
extern "C" void kernel_launch(void* const* d_in, const int* in_sizes, int n_in,
                              void* d_out, int out_size, void* d_ws,
                              size_t ws_size, hipStream_t stream) {
  (void)in_sizes; (void)n_in; (void)out_size; (void)ws_size;
  const float* x       = (const float*)d_in[0];
  const float* conv2_w = (const float*)d_in[1];
  const float* conv2_b = (const float*)d_in[2];
  const float* bn2_g   = (const float*)d_in[3];
  const float* bn2_b   = (const float*)d_in[4];
  const float* bn2_m   = (const float*)d_in[5];
  const float* bn2_v   = (const float*)d_in[6];
  const float* temper  = (const float*)d_in[7];
  const float* w1_w    = (const float*)d_in[8];
  const float* w1_b    = (const float*)d_in[9];
  const float* bnw_g   = (const float*)d_in[10];
  const float* bnw_b   = (const float*)d_in[11];
  const float* bnw_m   = (const float*)d_in[12];
  const float* bnw_v   = (const float*)d_in[13];
  const float* w2_w    = (const float*)d_in[14];
  const float* w2_b    = (const float*)d_in[15];
  const float* proj_w  = (const float*)d_in[16];
  const float* proj_b  = (const float*)d_in[17];
  float* out = (float*)d_out;

  const size_t NB = 8ull * 256 * 4096;       // one (b,256,64,64) f32 plane
  float* W = (float*)d_ws;
  float* A  = W + 0 * NB;                    // conv_out, later out_f2
  float* Bu = W + 1 * NB;
  float* Cu = W + 2 * NB;
  float* D  = W + 3 * NB;                    // f.re
  float* E  = W + 4 * NB;                    // f.im
  float* F  = W + 5 * NB;
  float* G  = W + 6 * NB;
  float* I  = W + 7 * NB;                    // out_l2
  float* g1buf = W + 8 * NB;                 // (8,16,4096)
  float* gramR = g1buf + 8 * 16 * 4096;
  float* gramI = gramR + 64 * 1024;
  float* attnR = gramI + 64 * 1024;
  float* attnI = attnR + 64 * 1024;
  float* norms = attnI + 64 * 1024;          // 2048
  float* wrep  = norms + 2048;               // 9*256*512 repacked conv weights

  // 0) weight repack; 1) conv + BN + ReLU -> A
  k_repack_w<<<4608, 256, 0, stream>>>(conv2_w, wrep);
  k_conv_bn_relu<<<2048, 512, 0, stream>>>(x, wrep, conv2_b, bn2_g, bn2_b,
                                           bn2_m, bn2_v, A);
  // 2) fft2 -> f in (D,E)
  k_dft64<<<2048, 256, 0, stream>>>(A, nullptr, Bu, Cu, 0, -1.f, 1.f);
  k_dft64<<<2048, 256, 0, stream>>>(Bu, Cu, D, E, 1, -1.f, 1.f);

  // 3) gating branch -> gated spectrum, ifft2, |.| -> I
  k_g1<<<256, 256, 0, stream>>>(D, w1_w, w1_b, bnw_g, bnw_b, bnw_m, bnw_v,
                                g1buf);
  k_gate<<<4096, 256, 0, stream>>>(g1buf, w2_w, w2_b, D, E, F, G);
  k_dft64<<<2048, 256, 0, stream>>>(F, G, Bu, Cu, 1, +1.f, 1.f / 64.f);
  k_dft64<<<2048, 256, 0, stream>>>(Bu, Cu, F, G, 0, +1.f, 1.f / 64.f);
  k_abs<<<32768, 256, 0, stream>>>(F, G, I);

  // 4) attention -> (F,G)
  k_rownorm<<<2048, 256, 0, stream>>>(D, E, norms);
  k_gram<<<64, 128, 0, stream>>>(D, E, gramR, gramI);
  k_softmax<<<64, dim3(32, 32), 0, stream>>>(gramR, gramI, norms, temper,
                                             attnR, attnI);
  k_attn_apply<<<4096, 256, 0, stream>>>(attnR, attnI, D, E, F, G);

  // 5) ifft2 over (32,4096): 32-pt IDFT + four-step 4096-pt IDFT
  k_idft32_left<<<4096, 256, 0, stream>>>(F, G, Bu, Cu);
  k_dft64<<<2048, 256, 0, stream>>>(Bu, Cu, D, E, 1, +1.f, 1.f / 64.f);
  k_twiddle4096<<<32768, 256, 0, stream>>>(D, E);
  k_dft64<<<2048, 256, 0, stream>>>(D, E, F, G, 0, +1.f, 1.f / 64.f);
  k_abs_remap<<<32768, 256, 0, stream>>>(F, G, A);   // out_f2 -> A

  // 6) concat + residual + projection -> d_out
  k_proj<<<8192, 256, 0, stream>>>(A, I, x, proj_w, proj_b, out);
}